// DeeperGCN_65369402245674
// MI455X (gfx1250) — compile-verified
//
#include <hip/hip_runtime.h>
#include <hip/hip_bf16.h>

typedef float v2f __attribute__((ext_vector_type(2)));
typedef float v8f __attribute__((ext_vector_type(8)));

#define NNODES 50000
#define NEDGES 1600000
#define DIM 128
#define DIM2 256
#define EPSC 1e-7f
#define LN_EPSC 1e-5f

// ---- helpers -------------------------------------------------------------

// Order-preserving float->uint map (monotone for all floats incl. sign)
__device__ __forceinline__ unsigned fmap(float f) {
  unsigned u = __float_as_uint(f);
  return (u & 0x80000000u) ? ~u : (u | 0x80000000u);
}
__device__ __forceinline__ float funmap(unsigned u) {
  unsigned v = (u & 0x80000000u) ? (u & 0x7FFFFFFFu) : ~u;
  return __uint_as_float(v);
}
__device__ __forceinline__ float wave_sum32(float v) {
  for (int off = 16; off > 0; off >>= 1) v += __shfl_xor(v, off, 32);
  return v;
}

// ---- edge pass 1: per-feature segment max of s = beta*msg ---------------
// one wave (32 lanes) per edge, each lane handles 4 features (float4)
__global__ void edge_max_kernel(const float* __restrict__ hn,
                                const int* __restrict__ ei,
                                const float* __restrict__ ea,
                                unsigned* __restrict__ m_u,
                                const float* __restrict__ beta, int layer) {
  long long wid = (long long)blockIdx.x * 8 + (threadIdx.x >> 5);
  if (wid >= NEDGES) return;
  int lane = threadIdx.x & 31;
  int src = ei[wid];
  int dst = ei[(long long)NEDGES + wid];
  float bt = beta[layer];
  const float4 xv = *(const float4*)(hn + (long long)src * DIM + lane * 4);
  const float4 av = *(const float4*)(ea + wid * (long long)DIM + lane * 4);
  float g0 = fmaxf(xv.x + av.x, 0.f) + EPSC;
  float g1 = fmaxf(xv.y + av.y, 0.f) + EPSC;
  float g2 = fmaxf(xv.z + av.z, 0.f) + EPSC;
  float g3 = fmaxf(xv.w + av.w, 0.f) + EPSC;
  unsigned* p = m_u + (long long)dst * DIM + lane * 4;
  atomicMax(p + 0, fmap(bt * g0));
  atomicMax(p + 1, fmap(bt * g1));
  atomicMax(p + 2, fmap(bt * g2));
  atomicMax(p + 3, fmap(bt * g3));
}

// ---- edge pass 2: e = exp(s - m[dst]); num += e*msg; den += e -----------
__global__ void edge_sum_kernel(const float* __restrict__ hn,
                                const int* __restrict__ ei,
                                const float* __restrict__ ea,
                                const unsigned* __restrict__ m_u,
                                float* __restrict__ num,
                                float* __restrict__ den,
                                const float* __restrict__ beta, int layer) {
  long long wid = (long long)blockIdx.x * 8 + (threadIdx.x >> 5);
  if (wid >= NEDGES) return;
  int lane = threadIdx.x & 31;
  int src = ei[wid];
  int dst = ei[(long long)NEDGES + wid];
  float bt = beta[layer];
  const float4 xv = *(const float4*)(hn + (long long)src * DIM + lane * 4);
  const float4 av = *(const float4*)(ea + wid * (long long)DIM + lane * 4);
  float g0 = fmaxf(xv.x + av.x, 0.f) + EPSC;
  float g1 = fmaxf(xv.y + av.y, 0.f) + EPSC;
  float g2 = fmaxf(xv.z + av.z, 0.f) + EPSC;
  float g3 = fmaxf(xv.w + av.w, 0.f) + EPSC;
  long long base = (long long)dst * DIM + lane * 4;
  float e0 = __expf(bt * g0 - funmap(m_u[base + 0]));
  float e1 = __expf(bt * g1 - funmap(m_u[base + 1]));
  float e2 = __expf(bt * g2 - funmap(m_u[base + 2]));
  float e3 = __expf(bt * g3 - funmap(m_u[base + 3]));
  atomicAdd(den + base + 0, e0);
  atomicAdd(den + base + 1, e1);
  atomicAdd(den + base + 2, e2);
  atomicAdd(den + base + 3, e3);
  atomicAdd(num + base + 0, e0 * g0);
  atomicAdd(num + base + 1, e1 * g1);
  atomicAdd(num + base + 2, e2 * g2);
  atomicAdd(num + base + 3, e3 * g3);
}

// ---- X = hn + aggr, aggr = num/den (0 for empty segments) ---------------
__global__ void aggr_kernel(const float* __restrict__ hn,
                            const float* __restrict__ num,
                            const float* __restrict__ den,
                            float* __restrict__ X) {
  long long i = ((long long)blockIdx.x * blockDim.x + threadIdx.x) * 4;
  if (i >= (long long)NNODES * DIM) return;
  float4 h4 = *(const float4*)(hn + i);
  float4 n4 = *(const float4*)(num + i);
  float4 d4 = *(const float4*)(den + i);
  float4 o;
  o.x = h4.x + (d4.x > 0.f ? n4.x / d4.x : 0.f);
  o.y = h4.y + (d4.y > 0.f ? n4.y / d4.y : 0.f);
  o.z = h4.z + (d4.z > 0.f ? n4.z / d4.z : 0.f);
  o.w = h4.w + (d4.w > 0.f ? n4.w / d4.w : 0.f);
  *(float4*)(X + i) = o;
}

// ---- GEMM: Out = [relu](X @ W + bias [+ res]) via V_WMMA_F32_16X16X4_F32
// W (K x Nw, 128KB) staged in LDS with padded stride (bank-conflict-free
// wmma B-fragment reads). Each wave computes a 16x64 output strip: one A
// fragment feeds 4 wmma issues per K-step.
__global__ void gemm_wmma_lds_kernel(const float* __restrict__ X,
                                     const float* __restrict__ W,
                                     const float* __restrict__ bias,
                                     const float* res, float* Out,
                                     int K, int Nw, int tiles, int colGroups,
                                     int nwShift, int relu_flag) {
  extern __shared__ float sW[];
  const int stride = Nw + 16;  // 2*stride % 64 == 32 -> halves hit disjoint banks
  const int tid = threadIdx.x;

  // cooperative stage of W[K x Nw] into LDS (float4 per thread per step)
  const int total = K << nwShift;  // K * Nw
  for (int i = tid * 4; i < total; i += 256 * 4) {
    int kk = i >> nwShift;
    int nn = i & (Nw - 1);
    *(float4*)(sW + kk * stride + nn) = *(const float4*)(W + i);
  }
  __syncthreads();

  int wid = blockIdx.x * 8 + (tid >> 5);
  if (wid >= tiles) return;
  int tm = wid / colGroups;
  int cg = wid - tm * colGroups;
  int m0 = tm * 16;
  int n0 = cg * 64;
  int lane = tid & 31;
  int r = lane & 15;
  int half = lane >> 4;

  // A fragment: lane r covers row m0+r; VGPR0/1 hold K = k+2*half+{0,1}
  const float* Arow = X + (size_t)(m0 + r) * K + 2 * half;
  // B fragments from LDS: lane r covers col n0+16t+r
  const float* B0 = sW + (size_t)(2 * half) * stride + n0 + r;

  v8f acc0 = {}, acc1 = {}, acc2 = {}, acc3 = {};
  for (int k = 0; k < K; k += 4) {
    v2f a = *(const v2f*)(Arow + k);
    const float* Bk = B0 + (size_t)k * stride;
    v2f b0, b1, b2, b3;
    b0.x = Bk[0];  b0.y = Bk[stride];
    b1.x = Bk[16]; b1.y = Bk[stride + 16];
    b2.x = Bk[32]; b2.y = Bk[stride + 32];
    b3.x = Bk[48]; b3.y = Bk[stride + 48];
    acc0 = __builtin_amdgcn_wmma_f32_16x16x4_f32(false, a, false, b0, (short)0,
                                                 acc0, false, false);
    acc1 = __builtin_amdgcn_wmma_f32_16x16x4_f32(false, a, false, b1, (short)0,
                                                 acc1, false, false);
    acc2 = __builtin_amdgcn_wmma_f32_16x16x4_f32(false, a, false, b2, (short)0,
                                                 acc2, false, false);
    acc3 = __builtin_amdgcn_wmma_f32_16x16x4_f32(false, a, false, b3, (short)0,
                                                 acc3, false, false);
  }

  auto epilogue = [&](const v8f& acc, int t) {
    float bv = bias[n0 + 16 * t + r];
#pragma unroll
    for (int j = 0; j < 8; ++j) {
      int row = m0 + j + 8 * half;
      size_t idx = (size_t)row * (size_t)Nw + n0 + 16 * t + r;
      float v = acc[j] + bv;
      if (res) v += res[idx];
      if (relu_flag) v = fmaxf(v, 0.f);
      Out[idx] = v;
    }
  };
  epilogue(acc0, 0);
  epilogue(acc1, 1);
  epilogue(acc2, 2);
  epilogue(acc3, 3);
}

// ---- hn = relu(layer_norm(h, scale[layer], bias[layer])) ----------------
// one wave per node, each lane handles 4 features
__global__ void ln_relu_kernel(const float* __restrict__ h,
                               const float* __restrict__ scale,
                               const float* __restrict__ bias,
                               float* __restrict__ hn, int layer) {
  int node = blockIdx.x * 8 + (threadIdx.x >> 5);
  if (node >= NNODES) return;
  int lane = threadIdx.x & 31;
  const float4 v = *(const float4*)(h + (long long)node * DIM + lane * 4);
  float s1 = v.x + v.y + v.z + v.w;
  float s2 = v.x * v.x + v.y * v.y + v.z * v.z + v.w * v.w;
  s1 = wave_sum32(s1);
  s2 = wave_sum32(s2);
  float mean = s1 * (1.f / DIM);
  float var = s2 * (1.f / DIM) - mean * mean;
  float rs = rsqrtf(var + LN_EPSC);
  const float* sc = scale + (size_t)layer * DIM + lane * 4;
  const float* bi = bias + (size_t)layer * DIM + lane * 4;
  float4 o;
  o.x = fmaxf((v.x - mean) * rs * sc[0] + bi[0], 0.f);
  o.y = fmaxf((v.y - mean) * rs * sc[1] + bi[1], 0.f);
  o.z = fmaxf((v.z - mean) * rs * sc[2] + bi[2], 0.f);
  o.w = fmaxf((v.w - mean) * rs * sc[3] + bi[3], 0.f);
  *(float4*)(hn + (long long)node * DIM + lane * 4) = o;
}

// ---- out = relu(layer_norm(h)) @ Wout + bout  -> [N,1] ------------------
__global__ void final_kernel(const float* __restrict__ h,
                             const float* __restrict__ scale,
                             const float* __restrict__ bias,
                             const float* __restrict__ Wout,
                             const float* __restrict__ bout,
                             float* __restrict__ out) {
  int node = blockIdx.x * 8 + (threadIdx.x >> 5);
  if (node >= NNODES) return;
  int lane = threadIdx.x & 31;
  const float4 v = *(const float4*)(h + (long long)node * DIM + lane * 4);
  float s1 = v.x + v.y + v.z + v.w;
  float s2 = v.x * v.x + v.y * v.y + v.z * v.z + v.w * v.w;
  s1 = wave_sum32(s1);
  s2 = wave_sum32(s2);
  float mean = s1 * (1.f / DIM);
  float var = s2 * (1.f / DIM) - mean * mean;
  float rs = rsqrtf(var + LN_EPSC);
  const float* sc = scale + lane * 4;
  const float* bi = bias + lane * 4;
  const float* wo = Wout + lane * 4;
  float t = 0.f;
  t += fmaxf((v.x - mean) * rs * sc[0] + bi[0], 0.f) * wo[0];
  t += fmaxf((v.y - mean) * rs * sc[1] + bi[1], 0.f) * wo[1];
  t += fmaxf((v.z - mean) * rs * sc[2] + bi[2], 0.f) * wo[2];
  t += fmaxf((v.w - mean) * rs * sc[3] + bi[3], 0.f) * wo[3];
  t = wave_sum32(t);
  if (lane == 0) out[node] = t + bout[0];
}

// ---- launcher ------------------------------------------------------------

extern "C" void kernel_launch(void* const* d_in, const int* in_sizes, int n_in,
                              void* d_out, int out_size, void* d_ws, size_t ws_size,
                              hipStream_t stream) {
  const float* x        = (const float*)d_in[0];
  const int*   ei       = (const int*)d_in[1];
  const float* ea       = (const float*)d_in[2];
  const float* ln_scale = (const float*)d_in[3];
  const float* ln_bias  = (const float*)d_in[4];
  const float* W1       = (const float*)d_in[5];
  const float* b1       = (const float*)d_in[6];
  const float* W2       = (const float*)d_in[7];
  const float* b2       = (const float*)d_in[8];
  const float* beta     = (const float*)d_in[9];
  const float* lnf_s    = (const float*)d_in[10];
  const float* lnf_b    = (const float*)d_in[11];
  const float* Wout     = (const float*)d_in[12];
  const float* bout     = (const float*)d_in[13];

  const size_t ND = (size_t)NNODES * DIM;
  float* base = (float*)d_ws;
  float* h_buf  = base;             // [N,D]   running residual state
  float* hn_buf = base + ND;        // [N,D]   conv input (relu(LN(h)))
  float* x_buf  = base + 2 * ND;    // [N,D]   m (as uint) then reused as X
  float* num    = base + 3 * ND;    // [N,D]
  float* den    = base + 4 * ND;    // [N,D]
  float* y_buf  = num;              // [N,2D]  overlays num+den after consumed
  unsigned* m_u = (unsigned*)x_buf;

  const int edgeBlocks = (NEDGES + 7) / 8;      // wave per edge, 8 waves/block
  const int nodeBlocks = (NNODES + 7) / 8;      // wave per node
  const int elemBlocks = (int)((ND / 4 + 255) / 256);

  for (int layer = 0; layer < 3; ++layer) {
    const float* hin;
    if (layer == 0) {
      hin = x;
    } else {
      ln_relu_kernel<<<nodeBlocks, 256, 0, stream>>>(h_buf, ln_scale, ln_bias,
                                                     hn_buf, layer);
      hin = hn_buf;
    }

    // zero m (uint 0 == mapped -inf), num, den in one contiguous memset
    hipMemsetAsync(x_buf, 0, 3 * ND * sizeof(float), stream);

    edge_max_kernel<<<edgeBlocks, 256, 0, stream>>>(hin, ei, ea, m_u, beta, layer);
    edge_sum_kernel<<<edgeBlocks, 256, 0, stream>>>(hin, ei, ea, m_u, num, den,
                                                    beta, layer);
    aggr_kernel<<<elemBlocks, 256, 0, stream>>>(hin, num, den, x_buf);

    // Y = relu(X @ W1 + b1)   [N,128] x [128,256]
    {
      int colGroups = DIM2 / 64;                      // 4
      int tiles = (NNODES / 16) * colGroups;          // 12500
      size_t ldsBytes = (size_t)DIM * (DIM2 + 16) * sizeof(float);  // 139264
      gemm_wmma_lds_kernel<<<(tiles + 7) / 8, 256, ldsBytes, stream>>>(
          x_buf, W1 + (size_t)layer * DIM * DIM2, b1 + (size_t)layer * DIM2,
          nullptr, y_buf, DIM, DIM2, tiles, colGroups, /*nwShift=*/8,
          /*relu=*/1);
    }
    // h = Y @ W2 + b2 (+ h_old for layers >= 1)   [N,256] x [256,128]
    {
      int colGroups = DIM / 64;                       // 2
      int tiles = (NNODES / 16) * colGroups;          // 6250
      size_t ldsBytes = (size_t)DIM2 * (DIM + 16) * sizeof(float);  // 147456
      gemm_wmma_lds_kernel<<<(tiles + 7) / 8, 256, ldsBytes, stream>>>(
          y_buf, W2 + (size_t)layer * DIM2 * DIM, b2 + (size_t)layer * DIM,
          (layer > 0) ? h_buf : nullptr, h_buf, DIM2, DIM, tiles, colGroups,
          /*nwShift=*/7, /*relu=*/0);
    }
  }

  final_kernel<<<nodeBlocks, 256, 0, stream>>>(h_buf, lnf_s, lnf_b, Wout, bout,
                                               (float*)d_out);
}